// regressor_ffa_46042049413677
// MI455X (gfx1250) — compile-verified
//
#include <hip/hip_runtime.h>
#include <math.h>
#include <stdint.h>

// ---------------------------------------------------------------------------
// FFA (fast folding) pipeline for MI455X (gfx1250, wave32).
//   out[b,c, cfg_slice] = (max - median)/std/sqrt(ini_rows) of folded profiles
// Memory-bound (33.5MB input, all buffers L2-resident). CDNA5 paths used:
//   - v_wmma_f32_16x16x4_f32 adder-tree for the std-dev reduction
//   - global_load_async_to_lds_b32 / global_store_async_from_lds_b32 for the
//     transposing LDS gather/scatter in the butterfly kernel (ASYNCcnt)
// ---------------------------------------------------------------------------

typedef __attribute__((ext_vector_type(2))) float v2f;
typedef __attribute__((ext_vector_type(8))) float v8f;

#define BC      32          // 8 * 4
#define N_ELEM  262144
#define OUT_ROW 81920

__device__ __forceinline__ unsigned lds_off32(const void* p) {
  // generic pointer to LDS: low 32 bits are the LDS byte offset
  return (unsigned)(size_t)p;
}

// ------------------------------ std-dev -------------------------------------
// One block per (b,c). 8 waves, each wave reduces a contiguous 32768-elem chunk
// 64 elements per WMMA issue: A (16x4 f32) x B(ones 4x16) + C accumulates row
// sums replicated over 16 columns; sum over all lanes/VGPRs = 16 * chunk sum.
__global__ __launch_bounds__(256) void k_std(const float* __restrict__ x,
                                             float* __restrict__ stdev) {
  const int bc   = blockIdx.x;
  const int wave = threadIdx.x >> 5;
  const int lane = threadIdx.x & 31;
  const float* row = x + (size_t)bc * N_ELEM;
  const int chunk = N_ELEM / 8;                       // 32768 per wave
  const float* wbase = row + wave * chunk;
  // lane -> A[m][k] slot: lanes 0-15 hold K=0,1 of row m=lane; 16-31 hold K=2,3
  const int loff = (lane & 15) * 4 + (lane >> 4) * 2;

  v8f acc_s = {};
  v8f acc_q = {};
  v2f ones; ones.x = 1.0f; ones.y = 1.0f;

  for (int it = 0; it < chunk / 64; ++it) {
    const float* p = wbase + it * 64 + loff;
    v2f a; a.x = p[0]; a.y = p[1];
    v2f q; q.x = a.x * a.x; q.y = a.y * a.y;
    acc_s = __builtin_amdgcn_wmma_f32_16x16x4_f32(false, a, false, ones,
                                                  (short)0, acc_s, false, false);
    acc_q = __builtin_amdgcn_wmma_f32_16x16x4_f32(false, q, false, ones,
                                                  (short)0, acc_q, false, false);
  }
  float s = 0.0f, qq = 0.0f;
  for (int v = 0; v < 8; ++v) { s += acc_s[v]; qq += acc_q[v]; }
  // wave32 all-lane reduce; every column holds full sum -> grand total = sum/16
  for (int off = 16; off > 0; off >>= 1) {
    s  += __shfl_xor(s,  off, 32);
    qq += __shfl_xor(qq, off, 32);
  }
  s *= (1.0f / 16.0f);
  qq *= (1.0f / 16.0f);

  __shared__ float ls[8], lq[8];
  if (lane == 0) { ls[wave] = s; lq[wave] = qq; }
  __syncthreads();
  if (threadIdx.x == 0) {
    float S = 0.0f, Q = 0.0f;
    for (int w = 0; w < 8; ++w) { S += ls[w]; Q += lq[w]; }
    const float n = (float)N_ELEM;
    float var = (Q - S * S / n) / (n - 1.0f);            // ddof = 1
    stdev[bc] = sqrtf(fmaxf(var, 0.0f));
  }
}

// ------------------------------ downsample ----------------------------------
// scale 2: down2[2j]   = .5*(x[4j]+x[4j+1]); down2[2j+1] = .5*(x[4j+2]+x[4j+3])
// scale 4: down4[j]    = .5*(x[4j+1]+x[4j+2])          (w = 0.5 exactly)
__global__ __launch_bounds__(256) void k_down(const float* __restrict__ x,
                                              float* __restrict__ d2,
                                              float* __restrict__ d4) {
  int idx = blockIdx.x * 256 + threadIdx.x;              // 32 * 65536 threads
  int j  = idx & 65535;
  int bc = idx >> 16;
  const float4 v = reinterpret_cast<const float4*>(x + (size_t)bc * N_ELEM)[j];
  float* p2 = d2 + (size_t)bc * 131072 + 2 * j;
  p2[0] = 0.5f * (v.x + v.y);
  p2[1] = 0.5f * (v.z + v.w);
  d4[(size_t)bc * 65536 + j] = 0.5f * (v.y + v.z);
}

// ------------------------------ FFA stages 1..9 in LDS -----------------------
// One workgroup = 512 consecutive folded rows of one (b,c). Async-gathers
// (trunc + zero-pad, transposing) from the downsample buffer straight into
// LDS, runs stages g=2..512 in LDS, async-scatters to global in [bc][col][row]
// layout (row contiguous) for coalesced global stages.
template <int BINS>
__global__ __launch_bounds__(256) void k_ffa_lds(const float* __restrict__ down,
                                                 float* __restrict__ bufA,
                                                 int Lsrc, int ini_rows, int Rtot) {
  __shared__ float tile[512 * BINS];
  const int grp = blockIdx.x;
  const int bc  = blockIdx.y;
  const int tid = threadIdx.x;
  const int r0  = grp * 512;

  // gather: element e (memory order r*BINS+c) -> tile[c*512 + r]
  const float* src = down + (size_t)bc * Lsrc + (size_t)r0 * BINS;
  const int limit = (ini_rows - r0) * BINS;              // multiple of BINS
  for (int e = tid; e < 512 * BINS; e += 256) {
    const int ti = (e % BINS) * 512 + (e / BINS);
    if (e < limit) {
      unsigned la = lds_off32(&tile[ti]);
      const float* ga = src + e;
      asm volatile("global_load_async_to_lds_b32 %0, %1, off"
                   :: "v"(la), "v"(ga) : "memory");
    } else {
      tile[ti] = 0.0f;                                   // zero-pad rows
    }
  }
  asm volatile("s_wait_asynccnt 0" ::: "memory");        // my LDS writes landed
  __syncthreads();                                       // everyone's landed

  float tmp[2 * BINS];
  for (int s = 1; s <= 9; ++s) {
    const int g = 1 << s;
    #pragma unroll
    for (int p = 0; p < 2 * BINS; ++p) {
      const int e = tid + 256 * p;
      const int c = e >> 9;
      const int r = e & 511;
      const int i  = r & (g - 1);
      const int base = r - i;
      const int iA = i >> 1;
      const int iB = iA + (g >> 1);
      const int Bs = (i + 1) >> 1;
      int cb = c - (Bs % BINS); if (cb < 0) cb += BINS;
      tmp[p] = tile[c * 512 + base + iA] + tile[cb * 512 + base + iB];
    }
    __syncthreads();                                     // all reads done
    #pragma unroll
    for (int p = 0; p < 2 * BINS; ++p) {
      const int e = tid + 256 * p;
      tile[e] = tmp[p];                                  // e == c*512 + r
    }
    __syncthreads();
  }

  // scatter LDS -> bufA[bc][c][r0+r], coalesced along r; S_ENDPGM's implicit
  // wait-idle guarantees completion before the next kernel on the stream.
  float* dst = bufA + (size_t)bc * BINS * Rtot + r0;
  for (int e = tid; e < 512 * BINS; e += 256) {
    float* gp = dst + (size_t)(e >> 9) * Rtot + (e & 511);
    unsigned la = lds_off32(&tile[e]);
    asm volatile("global_store_async_from_lds_b32 %0, %1, off"
                 :: "v"(gp), "v"(la) : "memory");
  }
}

// ------------------------------ FFA global stage -----------------------------
// out[bc][c][r] = in[bc][c][base+i/2] + in[bc][(c-(i+1)/2) mod nc][base+i/2+g/2]
__global__ __launch_bounds__(256) void k_ffa_stage(const float* __restrict__ in,
                                                   float* __restrict__ out,
                                                   int nc, int Rtot, int s) {
  int idx = blockIdx.x * 256 + threadIdx.x;
  int r     = idx & (Rtot - 1);                          // Rtot is 2^k
  int plane = idx / Rtot;                                // bc*nc + c
  if (plane >= BC * nc) return;
  int c  = plane % nc;
  int bc = plane / nc;
  const int g = 1 << s;
  const int i  = r & (g - 1);
  const int base = r - i;
  const int iA = i >> 1;
  const int iB = iA + (g >> 1);
  const int Bs = (i + 1) >> 1;
  int cb = c - (Bs % nc); if (cb < 0) cb += nc;
  const float* pin = in + (size_t)bc * nc * Rtot;
  out[(size_t)plane * Rtot + r] =
      pin[(size_t)c * Rtot + base + iA] + pin[(size_t)cb * Rtot + base + iB];
}

// --------------------- fused last stage + max/median/snr ---------------------
// Final stage has g = Rtot (base = 0, i = r): compute its BINS outputs for row
// r directly from the stage-(k-1) buffer, then reduce. Saves a full R*BINS
// buffer write + read per config.
template <int BINS>
__global__ __launch_bounds__(256) void k_final_fused(const float* __restrict__ in,
                                                     const float* __restrict__ stdev,
                                                     float* __restrict__ out,
                                                     int Rtot, float scale,
                                                     int out_off) {
  int idx = blockIdx.x * 256 + threadIdx.x;
  if (idx >= BC * Rtot) return;
  int r  = idx & (Rtot - 1);
  int bc = idx / Rtot;
  const int iA = r >> 1;
  const int iB = iA + (Rtot >> 1);
  const int Bs = (r + 1) >> 1;
  const int Bsm = Bs % BINS;
  const float* pin = in + (size_t)bc * BINS * Rtot;

  float v[BINS];
  float mx = -3.402823466e38f;
  #pragma unroll
  for (int c = 0; c < BINS; ++c) {
    int cb = c - Bsm; if (cb < 0) cb += BINS;
    float val = pin[(size_t)c * Rtot + iA] + pin[(size_t)cb * Rtot + iB];
    v[c] = val;
    mx = fmaxf(mx, val);
  }
  #pragma unroll
  for (int i = 1; i < BINS; ++i) {                       // insertion sort
    float key = v[i];
    int j = i - 1;
    while (j >= 0 && v[j] > key) { v[j + 1] = v[j]; --j; }
    v[j + 1] = key;
  }
  const float med = v[(BINS - 1) / 2];
  out[(size_t)bc * OUT_ROW + out_off + r] = (mx - med) / stdev[bc] * scale;
}

// ------------------------------ host driver ----------------------------------
template <int BINS>
static void run_cfg(const float* down, int Lsrc, int ini_rows, int k, int Rtot,
                    int out_off, float* bufA, float* bufB, const float* stdv,
                    float* out, hipStream_t stream) {
  dim3 grid(Rtot / 512, BC);
  k_ffa_lds<BINS><<<grid, 256, 0, stream>>>(down, bufA, Lsrc, ini_rows, Rtot);
  float* cur = bufA;
  float* nxt = bufB;
  for (int s = 10; s <= k - 1; ++s) {                    // last stage is fused
    int total = BC * BINS * Rtot;
    k_ffa_stage<<<(total + 255) / 256, 256, 0, stream>>>(cur, nxt, BINS, Rtot, s);
    float* t = cur; cur = nxt; nxt = t;
  }
  // k=14: stages 10..13 -> cur == bufA ; k=12: stages 10..11 -> cur == bufA
  float scale = 1.0f / sqrtf((float)ini_rows);
  k_final_fused<BINS><<<(BC * Rtot + 255) / 256, 256, 0, stream>>>(
      cur, stdv, out, Rtot, scale, out_off);
}

extern "C" void kernel_launch(void* const* d_in, const int* in_sizes, int n_in,
                              void* d_out, int out_size, void* d_ws, size_t ws_size,
                              hipStream_t stream) {
  const float* x = (const float*)d_in[0];
  float* out = (float*)d_out;
  float* ws  = (float*)d_ws;

  // workspace layout (floats)
  float* stdv  = ws;                                    // 32 (padded to 256)
  float* down2 = ws + 256;                              // 32 * 131072
  float* down4 = down2 + (size_t)BC * 131072;           // 32 * 65536
  float* bufA  = down4 + (size_t)BC * 65536;            // 32 * 16384 * 13 max
  float* bufB  = bufA + (size_t)BC * 16384 * 13;

  k_std<<<BC, 256, 0, stream>>>(x, stdv);
  k_down<<<(BC * 65536) / 256, 256, 0, stream>>>(x, down2, down4);

  // PLAN = [(2, 10, 14), (4, 16, 20)]; output slices concatenated on axis 2
  run_cfg<10>(down2, 131072, 13107, 14, 16384,     0, bufA, bufB, stdv, out, stream);
  run_cfg<11>(down2, 131072, 11915, 14, 16384, 16384, bufA, bufB, stdv, out, stream);
  run_cfg<12>(down2, 131072, 10922, 14, 16384, 32768, bufA, bufB, stdv, out, stream);
  run_cfg<13>(down2, 131072, 10082, 14, 16384, 49152, bufA, bufB, stdv, out, stream);
  run_cfg<16>(down4,  65536,  4096, 12,  4096, 65536, bufA, bufB, stdv, out, stream);
  run_cfg<17>(down4,  65536,  3855, 12,  4096, 69632, bufA, bufB, stdv, out, stream);
  run_cfg<18>(down4,  65536,  3640, 12,  4096, 73728, bufA, bufB, stdv, out, stream);
  run_cfg<19>(down4,  65536,  3449, 12,  4096, 77824, bufA, bufB, stdv, out, stream);
}